// GCNConv_2001454760208
// MI455X (gfx1250) — compile-verified
//
#include <hip/hip_runtime.h>

typedef __attribute__((ext_vector_type(2))) float v2f;
typedef __attribute__((ext_vector_type(8))) float v8f;

#define N_NODES 100000
#define N_EDGES 1600000
#define IN_DIM  128
#define OUT_DIM 64
#define ROW_TILES (N_NODES / 16)   // 6250, exact

// Padded pair-row stride: 80 pairs = 640 B = 160 banks ≡ 32 (mod 64), so the
// upper half-wave (K=khalf+2 -> next pair-row) hits banks 32..63 while the
// lower half-wave hits banks 0..31 -> conflict-free ds_load_b64.
#define W_PAIR_STRIDE 80
#define W_PAIR_ROWS   (IN_DIM / 2)   // 64

// ---------------------------------------------------------------------------
// Kernel 1: out[n][j] = bias[j]
// ---------------------------------------------------------------------------
__global__ void gcn_init_bias(const float* __restrict__ bias,
                              float* __restrict__ out, int n) {
    int i = blockIdx.x * blockDim.x + threadIdx.x;
    if (i < n) out[i] = bias[i & (OUT_DIM - 1)];
}

// ---------------------------------------------------------------------------
// Kernel 2: support = X @ W via V_WMMA_F32_16X16X4_F32 (fp32 tensor path).
// One wave -> 16 rows x 64 cols (4 x v8f accumulators). 8 waves/block.
// W staged in LDS as interleaved K-pairs so each B fragment is a single
// 8B-contiguous ds_load_b64 into an even-aligned VGPR pair (no repack movs).
//
// fp32 WMMA VGPR layouts (ISA 7.12.2):
//   A (16x4):  lanes 0-15 hold M=lane  {VGPR0:K=0, VGPR1:K=1}
//              lanes16-31 hold M=l-16  {VGPR0:K=2, VGPR1:K=3}
//   B (4x16):  VGPRg: lanes 0-15 -> K=g,   N=lane
//                     lanes16-31 -> K=g+2, N=lane-16
//   C/D(16x16):VGPRg: lanes 0-15 -> M=g,   N=lane
//                     lanes16-31 -> M=g+8, N=lane-16
// ---------------------------------------------------------------------------
__global__ __launch_bounds__(256) void gcn_gemm_wmma(
    const float* __restrict__ X, const float* __restrict__ W,
    float* __restrict__ S) {
    __shared__ v2f w_lds[W_PAIR_ROWS * W_PAIR_STRIDE];   // 40 KB

    // Cooperative load + pair-interleave of W: w_lds[k2][n] = {W[2k2][n], W[2k2+1][n]}
    for (int p = threadIdx.x; p < W_PAIR_ROWS * OUT_DIM; p += 256) {
        const int k2 = p >> 6;        // pair row 0..63
        const int n  = p & 63;
        v2f t;
        t.x = W[(2 * k2 + 0) * OUT_DIM + n];
        t.y = W[(2 * k2 + 1) * OUT_DIM + n];
        w_lds[k2 * W_PAIR_STRIDE + n] = t;
    }
    __syncthreads();

    const int wave = threadIdx.x >> 5;                 // 0..7
    const int lane = threadIdx.x & 31;
    const int tile = blockIdx.x * 8 + wave;            // 16-row tile id
    if (tile >= ROW_TILES) return;                     // wave-uniform guard

    const int row0  = tile * 16;
    const int m     = lane & 15;                       // M (A) / N-within-tile (B,C)
    const int khalf = (lane >> 4) * 2;                 // 0 or 2

    v8f acc[4] = {};                                   // 4 col-tiles of 16
    const float* arow = X + (size_t)(row0 + m) * IN_DIM;

    for (int k0 = 0; k0 < IN_DIM; k0 += 4) {
        // A fragment: two consecutive K values for this lane (8B-aligned b64 load)
        v2f a = *(const v2f*)(arow + k0 + khalf);
        // B fragments: one ds_load_b64 each, immediate offsets t*128 B
        const v2f* wk = w_lds + ((k0 + khalf) >> 1) * W_PAIR_STRIDE + m;
#pragma unroll
        for (int t = 0; t < 4; ++t) {
            v2f b = wk[t * 16];
            acc[t] = __builtin_amdgcn_wmma_f32_16x16x4_f32(
                /*neg_a=*/false, a, /*neg_b=*/false, b,
                /*c_mod=*/(short)0, acc[t],
                /*reuse_a=*/false, /*reuse_b=*/false);
        }
    }

    // Write back D: VGPR g -> row (g | g+8), col = t*16 + (lane&15)
    const int mhi = (lane >> 4) * 8;
#pragma unroll
    for (int t = 0; t < 4; ++t) {
#pragma unroll
        for (int g = 0; g < 8; ++g) {
            S[(size_t)(row0 + mhi + g) * OUT_DIM + t * 16 + m] = acc[t][g];
        }
    }
}

// ---------------------------------------------------------------------------
// Kernel 3: edge scatter. 16 threads per edge; each thread does one b128
// gather from support (L2-resident: 25.6 MB << 192 MB L2) + 4
// global_atomic_add_f32 into out. Edge lists are streamed once -> NT loads
// so they don't evict the reused support/out working set from L2.
// ---------------------------------------------------------------------------
__global__ __launch_bounds__(256) void gcn_scatter(
    const float* __restrict__ S, const float* __restrict__ vals,
    const int* __restrict__ row, const int* __restrict__ col,
    float* __restrict__ out) {
    long long t = (long long)blockIdx.x * 256 + threadIdx.x;
    long long e = t >> 4;
    if (e >= N_EDGES) return;
    const int j0 = (int)(t & 15) * 4;

    const float v = __builtin_nontemporal_load(vals + e);
    const int   c = __builtin_nontemporal_load(col + e);
    const int   r = __builtin_nontemporal_load(row + e);

    const float4 s = *(const float4*)(S + (size_t)c * OUT_DIM + j0);
    float* o = out + (size_t)r * OUT_DIM + j0;
    atomicAdd(o + 0, v * s.x);
    atomicAdd(o + 1, v * s.y);
    atomicAdd(o + 2, v * s.z);
    atomicAdd(o + 3, v * s.w);
}

// ---------------------------------------------------------------------------
extern "C" void kernel_launch(void* const* d_in, const int* in_sizes, int n_in,
                              void* d_out, int out_size, void* d_ws, size_t ws_size,
                              hipStream_t stream) {
    const float* x    = (const float*)d_in[0];
    const float* w    = (const float*)d_in[1];
    const float* bias = (const float*)d_in[2];
    const float* vals = (const float*)d_in[3];
    const int*   row  = (const int*)d_in[4];
    const int*   col  = (const int*)d_in[5];
    float* out     = (float*)d_out;
    float* support = (float*)d_ws;   // 100000*64*4 = 25.6 MB scratch

    // 1) out = bias (broadcast)
    {
        const int n = N_NODES * OUT_DIM;
        gcn_init_bias<<<(n + 255) / 256, 256, 0, stream>>>(bias, out, n);
    }
    // 2) support = X @ W  (WMMA fp32)
    {
        const int blocks = (ROW_TILES + 7) / 8;   // 8 waves/block, 1 tile/wave
        gcn_gemm_wmma<<<blocks, 256, 0, stream>>>(x, w, support);
    }
    // 3) scatter-add over edges
    {
        const long long threads = (long long)N_EDGES * 16;
        const int blocks = (int)((threads + 255) / 256);
        gcn_scatter<<<blocks, 256, 0, stream>>>(support, vals, row, col, out);
    }
}